// conv_88837103551313
// MI455X (gfx1250) — compile-verified
//
#include <hip/hip_runtime.h>
#include <stdint.h>

#define HH 50
#define WW 50
#define NPIX (HH * WW)      // 2500
#define PW 52               // padded row width (1-px zero halo)
#define PELEMS (PW * PW)    // 2704
#define NTHREADS 1024

typedef float v2f __attribute__((ext_vector_type(2)));

// ---- helpers -------------------------------------------------------------

// Force a wave-uniform float into an SGPR.
__device__ __forceinline__ float uniform_f(const float* p) {
  return __int_as_float(__builtin_amdgcn_readfirstlane(__float_as_int(*p)));
}
// Wave-uniform packed pair -> even-aligned SGPR pair (legal VOP3P scalar src).
__device__ __forceinline__ v2f uniform_pair(const float* p0, const float* p1) {
  v2f r;
  r.x = uniform_f(p0);
  r.y = uniform_f(p1);
  return r;
}

__device__ __forceinline__ float fast_rcp(float x) { return __builtin_amdgcn_rcpf(x); }
__device__ __forceinline__ float sigm(float x)     { return fast_rcp(1.0f + __expf(-x)); }
__device__ __forceinline__ float tanh_fast(float x) {
  // tanh(x) = 1 - 2/(exp(2x)+1); saturates correctly via exp->inf->rcp->0
  float e = __expf(2.0f * x);
  return 1.0f - 2.0f * fast_rcp(e + 1.0f);
}

// CDNA5 async global->LDS copy (ASYNCcnt-tracked). LDS byte offset is the low
// 32 bits of the generic shared address (flat-LDS aperture layout).
__device__ __forceinline__ void async_copy_b32(void* lds_dst, const float* gsrc) {
  unsigned lds_off = (unsigned)(uintptr_t)lds_dst;
  unsigned long long gaddr = (unsigned long long)(uintptr_t)gsrc;
  asm volatile("global_load_async_to_lds_b32 %0, %1, off"
               :: "v"(lds_off), "v"(gaddr) : "memory");
}

__device__ __forceinline__ void wait_async0() {
#if __has_builtin(__builtin_amdgcn_s_wait_asynccnt)
  __builtin_amdgcn_s_wait_asynccnt(0);
#else
  asm volatile("s_wait_asynccnt 0x0" ::: "memory");
#endif
}

// ---- persistent ConvLSTM scan kernel ------------------------------------
// One workgroup = 32 wave32s on one WGP. State in LDS/registers; x_t tiles
// double-buffered in LDS via async loads; one workgroup barrier per step.
// Gate math packed pairwise: each tap is v_pk_fma_f32 with an SGPR-pair
// weight operand (2 FMAs per issue slot). Single step body (no time unroll):
// keeps uniform live ranges inside the 106-SGPR budget -> no SGPR spills.

__global__ __launch_bounds__(NTHREADS, 1)
void convlstm_scan_kernel(const float* __restrict__ x,
                          const float* __restrict__ W,
                          const float* __restrict__ bias,
                          float* __restrict__ out,
                          int T)
{
  __shared__ float xs[2][PELEMS];   // padded x_t, double buffered
  __shared__ float hs[2][PELEMS];   // padded h,   double buffered

  const int tid = (int)threadIdx.x;

  // Weights -> uniform packed pairs (SGPR pairs).
  // W layout: W[g][ci][ky][kx], ci=0 pairs with x, ci=1 with h; idx=((g*2+ci)*9+j)
  v2f wx01[9], wx23[9], wh01[9], wh23[9];
#pragma unroll
  for (int j = 0; j < 9; ++j) {
    wx01[j] = uniform_pair(W + ((0 * 2 + 0) * 9 + j), W + ((1 * 2 + 0) * 9 + j));
    wx23[j] = uniform_pair(W + ((2 * 2 + 0) * 9 + j), W + ((3 * 2 + 0) * 9 + j));
    wh01[j] = uniform_pair(W + ((0 * 2 + 1) * 9 + j), W + ((1 * 2 + 1) * 9 + j));
    wh23[j] = uniform_pair(W + ((2 * 2 + 1) * 9 + j), W + ((3 * 2 + 1) * 9 + j));
  }
  const v2f b01 = uniform_pair(bias + 0, bias + 1);
  const v2f b23 = uniform_pair(bias + 2, bias + 3);

  // Zero LDS (halos stay zero forever; interiors get overwritten each step).
  for (int i = tid; i < PELEMS; i += NTHREADS) {
    xs[0][i] = 0.0f; xs[1][i] = 0.0f; hs[0][i] = 0.0f; hs[1][i] = 0.0f;
  }

  // Each thread owns <=3 pixels: tid, tid+1024, tid+2048.
  int np = 0, pix[3] = {0, 0, 0}, q[3] = {0, 0, 0};
#pragma unroll
  for (int k = 0; k < 3; ++k) {
    int p = tid + k * NTHREADS;
    if (p < NPIX) {
      int y = p / WW, xx = p - y * WW;
      pix[k] = p;
      q[k]   = (y + 1) * PW + (xx + 1);
      np     = k + 1;
    }
  }
  float cst[3]  = {0.0f, 0.0f, 0.0f};  // cell state
  float hval[3] = {0.0f, 0.0f, 0.0f};  // hidden state

  __syncthreads();

  // Preload x_0 into xs[0].
#pragma unroll
  for (int k = 0; k < 3; ++k)
    if (k < np) async_copy_b32(&xs[0][q[k]], x + pix[k]);
  wait_async0();
  __syncthreads();

  int cur = 0;
  const float* xnext = x + NPIX;     // strength-reduced prefetch pointer
  for (int t = 0; t < T; ++t) {
    const int nxt = cur ^ 1;

    // Prefetch x_{t+1} into the other buffer (overlaps with compute below).
    if (t + 1 < T) {
#pragma unroll
      for (int k = 0; k < 3; ++k)
        if (k < np) async_copy_b32(&xs[nxt][q[k]], xnext + pix[k]);
    }
    xnext += NPIX;

    const float* __restrict__ xc = xs[cur];
    const float* __restrict__ hc = hs[cur];

#pragma unroll
    for (int k = 0; k < 3; ++k) {
      if (k >= np) break;
      const int base = q[k];
      v2f a01 = b01, a23 = b23;
#pragma unroll
      for (int r = 0; r < 3; ++r) {
#pragma unroll
        for (int c2 = 0; c2 < 3; ++c2) {
          const int o = base + (r - 1) * PW + (c2 - 1);
          const int j = r * 3 + c2;
          const float xv = xc[o];
          const float hv = hc[o];
          const v2f xsp = {xv, xv};       // splat -> op_sel / dual-mov
          const v2f hsp = {hv, hv};
          a01 = __builtin_elementwise_fma(wx01[j], xsp, a01);  // v_pk_fma_f32
          a23 = __builtin_elementwise_fma(wx23[j], xsp, a23);
          a01 = __builtin_elementwise_fma(wh01[j], hsp, a01);
          a23 = __builtin_elementwise_fma(wh23[j], hsp, a23);
        }
      }
      const float ig = sigm(a01.x);       // gate order: i, f, o, g
      const float fg = sigm(a01.y);
      const float og = sigm(a23.x);
      const float gg = tanh_fast(a23.y);
      const float cn = fg * cst[k] + ig * gg;
      cst[k]  = cn;
      hval[k] = og * tanh_fast(cn);
    }

    // Publish h_{t+1} into the other h buffer (nobody reads it this step).
#pragma unroll
    for (int k = 0; k < 3; ++k)
      if (k < np) hs[nxt][q[k]] = hval[k];

    wait_async0();     // my async x_{t+1} writes landed in LDS
    __syncthreads();   // everyone's h writes + x prefetches visible
    cur = nxt;
  }

  // out = relu(final h), flat (1,50,50)
#pragma unroll
  for (int k = 0; k < 3; ++k)
    if (k < np) out[pix[k]] = fmaxf(hval[k], 0.0f);
}

// ---- launch --------------------------------------------------------------

extern "C" void kernel_launch(void* const* d_in, const int* in_sizes, int n_in,
                              void* d_out, int out_size, void* d_ws, size_t ws_size,
                              hipStream_t stream) {
  const float* x = (const float*)d_in[0];   // (1, T, 1, 50, 50) f32
  const float* W = (const float*)d_in[1];   // (4, 2, 3, 3)      f32
  const float* b = (const float*)d_in[2];   // (4,)              f32
  float* out = (float*)d_out;               // (1, 50, 50)       f32

  const int T = in_sizes[0] / NPIX;         // 8192
  convlstm_scan_kernel<<<1, NTHREADS, 0, stream>>>(x, W, b, out, T);
}